// GIN_68573447848158
// MI455X (gfx1250) — compile-verified
//
#include <hip/hip_runtime.h>
#include <hip/hip_bf16.h>

typedef __attribute__((ext_vector_type(2))) float v2f;
typedef __attribute__((ext_vector_type(8))) float v8f;

#define GIN_H 128
#define GIN_N 100000
#define GIN_E 1600000
#define GIN_G 1000
#define GIN_DEPTH 4

// ---------------------------------------------------------------------------
// out = relu( (A [+ A2]) @ W + bias [+ res] )
// A:[M,128] row-major f32, W:[128,128] row-major, bias:[128], res:[M,128]
// Block = 128 threads = 4 waves; wave w computes rows [16*(4*bid+w), +16).
// Matrix math: V_WMMA_F32_16X16X4_F32 (full fp32 — matches the f32 reference).
// W is staged through LDS in two 64-row halves (padded stride 132 to avoid
// bank conflicts), cooperatively loaded with coalesced float4s.
// ---------------------------------------------------------------------------
__global__ __launch_bounds__(128) void gin_gemm128(
    const float* __restrict__ A, const float* __restrict__ A2,
    const float* __restrict__ W, const float* __restrict__ bias,
    const float* __restrict__ res, float* __restrict__ out, int M)
{
  __shared__ float lw[64 * 132];

  const int tid  = threadIdx.x;
  const int wave = tid >> 5;
  const int lane = tid & 31;
  const int half = lane >> 4;   // which 16-lane half of the wave
  const int l16  = lane & 15;
  const int row0 = (blockIdx.x * 4 + wave) * 16;
  const bool active = (row0 < M);
  const int arow = active ? (row0 + l16) : 0;

  v8f acc[8];
#pragma unroll
  for (int t = 0; t < 8; ++t)
#pragma unroll
    for (int r = 0; r < 8; ++r) acc[t][r] = 0.0f;

  for (int kt = 0; kt < 2; ++kt) {
    __syncthreads();
    // Stage 64 K-rows of W into LDS (64*128 floats), fully coalesced.
    for (int i = tid; i < 64 * 32; i += 128) {
      const int r  = i >> 5;
      const int c4 = (i & 31) << 2;
      const float4 v =
          *reinterpret_cast<const float4*>(&W[(size_t)(kt * 64 + r) * GIN_H + c4]);
      float* dp = &lw[r * 132 + c4];
      dp[0] = v.x; dp[1] = v.y; dp[2] = v.z; dp[3] = v.w;
    }
    __syncthreads();

    for (int kb = 0; kb < 64; kb += 4) {
      // A-matrix 16x4 f32 layout: lanes 0-15 carry K={0,1}, lanes 16-31 K={2,3}
      const int kloc = kb + 2 * half;
      const int kglb = kt * 64 + kloc;
      v2f a;
      const float* ap = &A[(size_t)arow * GIN_H + kglb];
      a.x = ap[0];
      a.y = ap[1];
      if (A2) {  // fused z = h + agg
        const float* ap2 = &A2[(size_t)arow * GIN_H + kglb];
        a.x += ap2[0];
        a.y += ap2[1];
      }
#pragma unroll
      for (int t = 0; t < 8; ++t) {
        const int col = t * 16 + l16;
        // B-matrix 4x16 layout mirrors A: lanes 0-15 rows K={0,1}, 16-31 K={2,3}
        v2f b;
        b.x = lw[kloc * 132 + col];
        b.y = lw[(kloc + 1) * 132 + col];
        acc[t] = __builtin_amdgcn_wmma_f32_16x16x4_f32(
            false, a, false, b, (short)0, acc[t], false, false);
      }
    }
  }

  if (!active) return;
  // C/D 16x16 f32 layout: VGPR r -> row (r + 8*half), col = lane%16 (+16*t)
#pragma unroll
  for (int t = 0; t < 8; ++t) {
    const int col = t * 16 + l16;
    const float bv = bias[col];
#pragma unroll
    for (int r = 0; r < 8; ++r) {
      const int row = row0 + r + 8 * half;
      float v = acc[t][r] + bv;
      if (res) v += res[(size_t)row * GIN_H + col];  // fused residual (h0)
      out[(size_t)row * GIN_H + col] = fmaxf(v, 0.0f);
    }
  }
}

// ---------------------------------------------------------------------------
// agg[dst[e]] += h[src[e]]  — 32 lanes per edge, float4 gather + 4 f32 atomics
// ---------------------------------------------------------------------------
__global__ __launch_bounds__(256) void gin_scatter(
    const int* __restrict__ src, const int* __restrict__ dst,
    const float* __restrict__ h, float* __restrict__ agg, int nE)
{
  const long long gid = (long long)blockIdx.x * 256 + threadIdx.x;
  const int e = (int)(gid >> 5);
  if (e >= nE) return;
  const int lane = (int)gid & 31;
  const int s = src[e];
  const int d = dst[e];
  const float4 v =
      *reinterpret_cast<const float4*>(&h[(size_t)s * GIN_H + lane * 4]);
  float* p = &agg[(size_t)d * GIN_H + lane * 4];
  atomicAdd(p + 0, v.x);
  atomicAdd(p + 1, v.y);
  atomicAdd(p + 2, v.z);
  atomicAdd(p + 3, v.w);
}

// pooled[batch[n]] += h[n]
__global__ __launch_bounds__(256) void gin_pool(
    const float* __restrict__ h, const int* __restrict__ batch,
    float* __restrict__ pooled, int n)
{
  const long long gid = (long long)blockIdx.x * 256 + threadIdx.x;
  const int node = (int)(gid >> 5);
  if (node >= n) return;
  const int lane = (int)gid & 31;
  const int g = batch[node];
  const float4 v =
      *reinterpret_cast<const float4*>(&h[(size_t)node * GIN_H + lane * 4]);
  float* p = &pooled[(size_t)g * GIN_H + lane * 4];
  atomicAdd(p + 0, v.x);
  atomicAdd(p + 1, v.y);
  atomicAdd(p + 2, v.z);
  atomicAdd(p + 3, v.w);
}

__global__ __launch_bounds__(256) void gin_fill0(float4* __restrict__ p, int n4)
{
  const int i = blockIdx.x * 256 + threadIdx.x;
  if (i < n4) p[i] = make_float4(0.f, 0.f, 0.f, 0.f);
}

// out[g] = dot(pooled[g], Wf) + bf  — one wave32 per graph, shuffle reduce
__global__ __launch_bounds__(256) void gin_readout(
    const float* __restrict__ pooled, const float* __restrict__ Wf,
    const float* __restrict__ bf, float* __restrict__ out, int G)
{
  const int gid = blockIdx.x * 256 + threadIdx.x;
  const int g = gid >> 5;
  if (g >= G) return;
  const int lane = gid & 31;
  float s = 0.f;
#pragma unroll
  for (int c = lane; c < GIN_H; c += 32) s += pooled[(size_t)g * GIN_H + c] * Wf[c];
#pragma unroll
  for (int off = 16; off > 0; off >>= 1) s += __shfl_down(s, off, 32);
  if (lane == 0) out[g] = s + bf[0];
}

extern "C" void kernel_launch(void* const* d_in, const int* in_sizes, int n_in,
                              void* d_out, int out_size, void* d_ws, size_t ws_size,
                              hipStream_t stream) {
  const float* x     = (const float*)d_in[0];
  const int*   eidx  = (const int*)d_in[1];
  // d_in[2] = edge_attr: read by the reference but unused in the math
  const int*   batch = (const int*)d_in[3];
  const float* W0 = (const float*)d_in[4];
  const float* b0 = (const float*)d_in[5];
  const float* W1 = (const float*)d_in[6];
  const float* b1 = (const float*)d_in[7];
  const float* W2 = (const float*)d_in[8];
  const float* b2 = (const float*)d_in[9];
  const float* Wf = (const float*)d_in[10];
  const float* bf = (const float*)d_in[11];
  float* out = (float*)d_out;

  const int N = GIN_N, E = GIN_E, G = GIN_G;
  const int* src = eidx;       // edge_index[0]
  const int* dst = eidx + E;   // edge_index[1]

  const size_t nh = (size_t)N * GIN_H;
  float* h0     = (float*)d_ws;
  float* h      = h0 + nh;
  float* agg    = h + nh;
  float* tmp    = agg + nh;
  float* pooled = tmp + nh;

  const int mtiles     = (N + 15) / 16;        // 6250
  const int gemmBlocks = (mtiles + 3) / 4;     // 1563
  const int aggN4      = N * (GIN_H / 4);      // float4 count for [N,H]
  const int scatBlocks = (int)(((long long)E * 32 + 255) / 256);
  const int poolBlocks = (int)(((long long)N * 32 + 255) / 256);

  // h0 = relu(x @ W0 + b0)
  gin_gemm128<<<gemmBlocks, 128, 0, stream>>>(x, nullptr, W0, b0, nullptr, h0, N);

  for (int i = 0; i < GIN_DEPTH; ++i) {
    const float* hin = (i == 0) ? h0 : h;  // h starts as h0; avoid a copy
    gin_fill0<<<(aggN4 + 255) / 256, 256, 0, stream>>>((float4*)agg, aggN4);
    gin_scatter<<<scatBlocks, 256, 0, stream>>>(src, dst, hin, agg, E);
    // tmp = relu((h + agg) @ W1[i] + b1[i])
    gin_gemm128<<<gemmBlocks, 128, 0, stream>>>(
        hin, agg, W1 + (size_t)i * GIN_H * GIN_H, b1 + (size_t)i * GIN_H,
        nullptr, tmp, N);
    // h = relu(tmp @ W2[i] + b2[i] + h0)
    gin_gemm128<<<gemmBlocks, 128, 0, stream>>>(
        tmp, nullptr, W2 + (size_t)i * GIN_H * GIN_H, b2 + (size_t)i * GIN_H,
        h0, h, N);
  }

  const int poolN4 = G * (GIN_H / 4);
  gin_fill0<<<(poolN4 + 255) / 256, 256, 0, stream>>>((float4*)pooled, poolN4);
  gin_pool<<<poolBlocks, 256, 0, stream>>>(h, batch, pooled, N);
  gin_readout<<<(G * 32 + 255) / 256, 256, 0, stream>>>(pooled, Wf, bf, out, G);

  (void)in_sizes; (void)n_in; (void)out_size; (void)ws_size;
}